// ConeProj3D_29850022707687
// MI455X (gfx1250) — compile-verified
//
#include <hip/hip_runtime.h>
#include <hip/hip_bf16.h>
#include <math.h>

// Cone-beam forward projection, MI455X (gfx1250, wave32).
// One lane per detector ray; 192-step ray march with trilinear gathers.
// Volume (67 MB) is L2-resident on MI455X (192 MB L2) -> gather-issue bound.

#define VOLN   256
#define N_ANG  16
#define DET_V  128
#define DET_U  128
#define N_STEP 192

// Per-angle sin/cos computed on the host (f32-rounded theta, libm-accurate
// sin/cos) and passed by value -> scalar s_loads from the kernarg segment.
struct AngleTrig {
  float s[N_ANG];
  float c[N_ANG];
};

__global__ __launch_bounds__(256) void ConeProj3D_29850022707687_kernel(
    const float* __restrict__ vol, float* __restrict__ out, AngleTrig trig) {
  const float DSI_  = 400.0f;
  const float DSD_  = 800.0f;
  const float DETSP = 5.0f;
  const float HALF  = 127.5f;
  // radius = sqrt(3)*128 = 221.70250336881629 (f64), matching jnp.float32 casts:
  const float T0 = 178.29749663118371f;   // float32(DSI - radius)
  const float DT = 2.3094010767585034f;   // float32(2*radius/192)

  int tid = blockIdx.x * blockDim.x + threadIdx.x;
  int u = tid & (DET_U - 1);
  int v = (tid >> 7) & (DET_V - 1);
  int a = blockIdx.x >> 6;          // 64 blocks per angle -> SGPR-uniform

  float sn = trig.s[a];
  float cs = trig.c[a];

  float uoff = ((float)u - 63.5f) * DETSP;
  float voff = ((float)v - 63.5f) * DETSP;

  // world coords ordered (z, y, x) to match volume index order
  float sz = 0.0f, sy = DSI_ * sn, sx = DSI_ * cs;            // source
  float dtz = voff;                                            // detector point
  float dty = -(DSD_ - DSI_) * sn + uoff * cs;
  float dtx = -(DSD_ - DSI_) * cs - uoff * sn;
  float ddz = dtz - sz, ddy = dty - sy, ddx = dtx - sx;
  float invn = 1.0f / sqrtf(ddz * ddz + ddy * ddy + ddx * ddx);
  float rz = ddz * invn, ry = ddy * invn, rx = ddx * invn;     // unit ray dir

  // Slab-clip against the open box (-128.5, 128.5)^3 in world coords:
  // samples outside it contribute exactly zero to the trilerp sum.
  float tmin = -3.0e38f, tmax = 3.0e38f;
  bool hit = true;
  {
    float o[3] = {sz, sy, sx};
    float d[3] = {rz, ry, rx};
#pragma unroll
    for (int k = 0; k < 3; ++k) {
      if (fabsf(d[k]) > 1e-7f) {
        float ta = (-128.5f - o[k]) / d[k];
        float tb = ( 128.5f - o[k]) / d[k];
        tmin = fmaxf(tmin, fminf(ta, tb));
        tmax = fminf(tmax, fmaxf(ta, tb));
      } else if (o[k] <= -128.5f || o[k] >= 128.5f) {
        hit = false;
      }
    }
  }
  int ilo = 0, ihi = -1;
  if (hit && tmax > tmin) {
    float flo = fminf(fmaxf((tmin - T0) / DT - 0.5f, -2.0f), (float)N_STEP + 2.0f);
    float fhi = fminf(fmaxf((tmax - T0) / DT - 0.5f, -2.0f), (float)N_STEP + 2.0f);
    ilo = (int)ceilf(flo) - 1;  if (ilo < 0) ilo = 0;              // +/-1 step margin;
    ihi = (int)floorf(fhi) + 1; if (ihi > N_STEP - 1) ihi = N_STEP - 1; // edges masked below
  }

  // Incremental marching: start position at step ilo, fixed per-step delta.
  float tstart = T0 + ((float)ilo + 0.5f) * DT;
  float pz = sz + tstart * rz + HALF;
  float py = sy + tstart * ry + HALF;
  float px = sx + tstart * rx + HALF;
  float stz = DT * rz, sty = DT * ry, stx = DT * rx;

  // Prefetch ~8 steps ahead along the ray (voxel-index delta, precomputed).
  int pfd = ((int)(8.0f * stz)) * 65536 +
            ((int)(8.0f * sty)) * 256 +
            ((int)(8.0f * stx));

  float acc = 0.0f;
  for (int i = ilo; i <= ihi; ++i) {
    float zf = floorf(pz), yf = floorf(py), xf = floorf(px);
    float fz = pz - zf, fy = py - yf, fx = px - xf;
    int iz = (int)zf, iy = (int)yf, ix = (int)xf;

    if (((unsigned)iz < 255u) & ((unsigned)iy < 255u) & ((unsigned)ix < 255u)) {
      // Fast interior path: 8 loads off one base address (immediate offsets).
      int base = (iz << 16) + (iy << 8) + ix;
      int pfi = base + pfd;                       // clamped speculative prefetch
      pfi = pfi < 0 ? 0 : pfi;
      pfi = pfi > (VOLN * VOLN * VOLN - 1) ? (VOLN * VOLN * VOLN - 1) : pfi;
      __builtin_prefetch(vol + pfi, 0, 0);        // -> global_prefetch_b8

      const float* p0 = vol + base;
      float v000 = p0[0],     v001 = p0[1];
      float v010 = p0[256],   v011 = p0[257];
      float v100 = p0[65536], v101 = p0[65537];
      float v110 = p0[65792], v111 = p0[65793];
      float r00 = v000 + fx * (v001 - v000);
      float r01 = v010 + fx * (v011 - v010);
      float r10 = v100 + fx * (v101 - v100);
      float r11 = v110 + fx * (v111 - v110);
      float r0  = r00 + fy * (r01 - r00);
      float r1  = r10 + fy * (r11 - r10);
      acc += r0 + fz * (r1 - r0);
    } else {
      // Boundary path: per-corner in-bounds masking (matches reference zero-pad).
      float part = 0.0f;
#pragma unroll
      for (int dcode = 0; dcode < 8; ++dcode) {
        int dz2 = dcode >> 2, dy2 = (dcode >> 1) & 1, dx2 = dcode & 1;
        int cz = iz + dz2, cy = iy + dy2, cx = ix + dx2;
        float w = (dz2 ? fz : 1.0f - fz) * (dy2 ? fy : 1.0f - fy) * (dx2 ? fx : 1.0f - fx);
        bool inb = ((unsigned)cz < 256u) && ((unsigned)cy < 256u) && ((unsigned)cx < 256u);
        int ccz = cz < 0 ? 0 : (cz > 255 ? 255 : cz);
        int ccy = cy < 0 ? 0 : (cy > 255 ? 255 : cy);
        int ccx = cx < 0 ? 0 : (cx > 255 ? 255 : cx);
        float vv = vol[(ccz << 16) + (ccy << 8) + ccx];
        part += (inb ? w : 0.0f) * vv;
      }
      acc += part;
    }

    pz += stz; py += sty; px += stx;   // incremental step
  }

  // Write-once sinogram: non-temporal store so it doesn't evict volume lines.
  __builtin_nontemporal_store(acc * DT, &out[tid]);   // [A,DV,DU] == [A*DV, DU]
}

extern "C" void kernel_launch(void* const* d_in, const int* in_sizes, int n_in,
                              void* d_out, int out_size, void* d_ws, size_t ws_size,
                              hipStream_t stream) {
  const float* vol = (const float*)d_in[0];   // [1,256,256,256,1] f32
  float* out = (float*)d_out;                 // [1, 16*128, 128, 1] f32

  // Host-side trig table: theta rounded exactly like the f32 reference
  // ((float)(2*pi/16) * k, f32 multiply), then libm sin/cos.
  AngleTrig trig;
  const float dtheta = 0.39269908169872414f;  // rounds to 0x3EC90FDB, same as jnp
  for (int k = 0; k < N_ANG; ++k) {
    float th = dtheta * (float)k;
    trig.s[k] = sinf(th);
    trig.c[k] = cosf(th);
  }

  const int total = N_ANG * DET_V * DET_U;    // 262144 rays, 1 lane each
  ConeProj3D_29850022707687_kernel<<<total / 256, 256, 0, stream>>>(vol, out, trig);
  (void)in_sizes; (void)n_in; (void)out_size; (void)d_ws; (void)ws_size;
}